// GGNNRelReason_21680994910744
// MI455X (gfx1250) — compile-verified
//
#include <hip/hip_runtime.h>
#include <hip/hip_bf16.h>

typedef float v2f __attribute__((ext_vector_type(2)));
typedef float v4f __attribute__((ext_vector_type(4)));
typedef float v8f __attribute__((ext_vector_type(8)));

#define NUM_OBJ     256
#define NUM_REL     4096
#define OBJ_DIM     4096   // == REL_DIM == K
#define HIDDEN      512
#define NUM_OBJ_CLS 151
#define NUM_REL_CLS 51

// ---------------------------------------------------------------------------
// 1) obj_dists2 = obj_logits (pass-through copy)
// ---------------------------------------------------------------------------
__global__ void copy_logits_kernel(const float* __restrict__ src,
                                   float* __restrict__ dst, int n) {
  int i = blockIdx.x * blockDim.x + threadIdx.x;
  if (i < n) dst[i] = src[i];
}

// ---------------------------------------------------------------------------
// 2) obj_preds = argmax(logits[:,1:]) + 1   (one thread per row; 150 scans)
// ---------------------------------------------------------------------------
__global__ void argmax_kernel(const float* __restrict__ logits,
                              float* __restrict__ preds_f,
                              int* __restrict__ preds_i) {
  int row = blockIdx.x * blockDim.x + threadIdx.x;
  if (row >= NUM_OBJ) return;
  const float* p = logits + row * NUM_OBJ_CLS;
  float best = p[1];
  int   bi   = 1;                  // class index already includes the +1
  for (int c = 2; c < NUM_OBJ_CLS; ++c) {
    float v = p[c];
    if (v > best) { best = v; bi = c; }   // '>' keeps first max, like jnp.argmax
  }
  preds_f[row] = (float)bi;
  preds_i[row] = bi;
}

// ---------------------------------------------------------------------------
// 3) sub_obj_preds gather
// ---------------------------------------------------------------------------
__global__ void gather_preds_kernel(const int* __restrict__ rel_inds,
                                    const int* __restrict__ preds_i,
                                    float* __restrict__ out2) {
  int r = blockIdx.x * blockDim.x + threadIdx.x;
  if (r >= NUM_REL) return;
  int s = rel_inds[r * 3 + 1];
  int o = rel_inds[r * 3 + 2];
  out2[r * 2 + 0] = (float)preds_i[s];
  out2[r * 2 + 1] = (float)preds_i[o];
}

// ---------------------------------------------------------------------------
// 4/5) C[M x 512] = A[M x 4096] * W[512 x 4096]^T + bias   (fp32 WMMA)
//
// Workgroup: 256 threads = 8 waves, computes a 64(M) x 128(N) block of C.
// Waves tiled 2(M) x 4(N); each wave owns 32x32 = 2x2 WMMA 16x16 tiles.
// K is staged through LDS in chunks of 32.
//
// Fragment layouts per CDNA5 ISA (CDNA5 ISA):
//   A 16x4 f32 : VGPR j, lane-half h  holds K = 2*h + j, M = lane%16
//   B 4x16 f32 : mirrored (K = 2*h + j, N = lane%16)
//   C/D 16x16  : VGPR v holds M = v + 8*half, N = lane%16
// ---------------------------------------------------------------------------
#define TK  32
#define LDP (TK + 4)   // 36 floats = 144B row stride: 16B aligned, conflict-free

__global__ __launch_bounds__(256) void gemm_bias_wmma(
    const float* __restrict__ A,     // M x 4096, row-major
    const float* __restrict__ W,     // 512 x 4096, row-major
    const float* __restrict__ bias,  // 512
    float* __restrict__ C) {         // M x 512, row-major
  __shared__ __align__(16) float As[64 * LDP];
  __shared__ __align__(16) float Bs[128 * LDP];

  const int tid  = threadIdx.x;
  const int lane = tid & 31;
  const int wave = tid >> 5;
  const int half = lane >> 4;   // 0|1 : selects K pair / M+8
  const int l16  = lane & 15;
  const int wm   = wave >> 2;   // 0..1 wave row
  const int wn   = wave & 3;    // 0..3 wave col
  const int m_blk = blockIdx.y * 64;
  const int n_blk = blockIdx.x * 128;
  const int K = OBJ_DIM;

  v8f zero = {};
  v8f acc[2][2];
  acc[0][0] = zero; acc[0][1] = zero; acc[1][0] = zero; acc[1][1] = zero;

  const int ldr = tid >> 3;        // 0..31  (row within a 32-row slab)
  const int ldc = (tid & 7) * 4;   // 0..28  (16B column offset)

  for (int k0 = 0; k0 < K; k0 += TK) {
    __syncthreads();
    // Stage A: 64 rows x 32 cols (b128 per thread)
    #pragma unroll
    for (int rr = 0; rr < 64; rr += 32) {
      v4f v = *(const v4f*)(A + (size_t)(m_blk + ldr + rr) * K + k0 + ldc);
      *(v4f*)(&As[(ldr + rr) * LDP + ldc]) = v;
    }
    // Stage B (weight rows): 128 rows x 32 cols
    #pragma unroll
    for (int rr = 0; rr < 128; rr += 32) {
      v4f v = *(const v4f*)(W + (size_t)(n_blk + ldr + rr) * K + k0 + ldc);
      *(v4f*)(&Bs[(ldr + rr) * LDP + ldc]) = v;
    }
    __syncthreads();

    #pragma unroll
    for (int kk = 0; kk < TK; kk += 4) {
      v2f a[2], b[2];
      #pragma unroll
      for (int i = 0; i < 2; ++i) {
        const float* ap = &As[(wm * 32 + i * 16 + l16) * LDP + kk + 2 * half];
        a[i][0] = ap[0];
        a[i][1] = ap[1];
      }
      #pragma unroll
      for (int j = 0; j < 2; ++j) {
        const float* bp = &Bs[(wn * 32 + j * 16 + l16) * LDP + kk + 2 * half];
        b[j][0] = bp[0];
        b[j][1] = bp[1];
      }
      #pragma unroll
      for (int i = 0; i < 2; ++i)
        #pragma unroll
        for (int j = 0; j < 2; ++j)
          acc[i][j] = __builtin_amdgcn_wmma_f32_16x16x4_f32(
              /*neg_a=*/false, a[i], /*neg_b=*/false, b[j],
              /*c_mod=*/(short)0, acc[i][j],
              /*reuse_a=*/false, /*reuse_b=*/false);
    }
  }

  // Epilogue: D layout -> global, fused bias add
  #pragma unroll
  for (int i = 0; i < 2; ++i) {
    #pragma unroll
    for (int j = 0; j < 2; ++j) {
      const int col = n_blk + wn * 32 + j * 16 + l16;
      const float bv = bias[col];
      float tmp[8];
      *(v8f*)tmp = acc[i][j];
      #pragma unroll
      for (int v = 0; v < 8; ++v) {
        const int row = m_blk + wm * 32 + i * 16 + 8 * half + v;
        C[(size_t)row * HIDDEN + col] = tmp[v] + bv;
      }
    }
  }
}

// ---------------------------------------------------------------------------
// 6) input_ggnn[r, c, :]: c=0 -> obj_h[sub], c=1 -> obj_h[obj], c>=2 -> vr_h[r]
//    445 MB write-once -> 128-bit non-temporal stores. vr_h row hits L2.
// ---------------------------------------------------------------------------
__global__ __launch_bounds__(256) void assemble_kernel(
    const float* __restrict__ objh, const float* __restrict__ vrh,
    const int* __restrict__ rel_inds, float* __restrict__ out) {
  const int r = blockIdx.x;
  const int s = rel_inds[r * 3 + 1];
  const int o = rel_inds[r * 3 + 2];
  const v4f* rowS = (const v4f*)(objh + (size_t)s * HIDDEN);
  const v4f* rowO = (const v4f*)(objh + (size_t)o * HIDDEN);
  const v4f* rowV = (const v4f*)(vrh + (size_t)r * HIDDEN);
  v4f* dst = (v4f*)(out + (size_t)r * (NUM_REL_CLS + 2) * HIDDEN);

  const int QROW = HIDDEN / 4;  // 128 x v4f per 512-float row
  for (int idx = threadIdx.x; idx < (NUM_REL_CLS + 2) * QROW; idx += blockDim.x) {
    const int c = idx / QROW;
    const int q = idx - c * QROW;
    v4f v = (c == 0) ? rowS[q] : (c == 1) ? rowO[q] : rowV[q];
    __builtin_nontemporal_store(v, &dst[idx]);
  }
}

// ---------------------------------------------------------------------------
extern "C" void kernel_launch(void* const* d_in, const int* in_sizes, int n_in,
                              void* d_out, int out_size, void* d_ws, size_t ws_size,
                              hipStream_t stream) {
  const float* obj_fmaps  = (const float*)d_in[0];
  const float* obj_logits = (const float*)d_in[1];
  const int*   rel_inds   = (const int*)d_in[2];
  const float* vr         = (const float*)d_in[3];
  const float* Wo         = (const float*)d_in[4];
  const float* bo         = (const float*)d_in[5];
  const float* Wr         = (const float*)d_in[6];
  const float* br         = (const float*)d_in[7];

  float* out = (float*)d_out;
  float* out_logits = out;                                              // 256*151
  float* out_preds  = out + NUM_OBJ * NUM_OBJ_CLS;                      // 256
  float* out_sop    = out_preds + NUM_OBJ;                              // 4096*2
  float* out_ggnn   = out_sop + NUM_REL * 2;                            // 4096*53*512

  float* objh    = (float*)d_ws;                        // 256*512
  float* vrh     = objh + NUM_OBJ * HIDDEN;             // 4096*512
  int*   preds_i = (int*)(vrh + NUM_REL * HIDDEN);      // 256 ints

  copy_logits_kernel<<<(NUM_OBJ * NUM_OBJ_CLS + 255) / 256, 256, 0, stream>>>(
      obj_logits, out_logits, NUM_OBJ * NUM_OBJ_CLS);
  argmax_kernel<<<1, 256, 0, stream>>>(obj_logits, out_preds, preds_i);
  gather_preds_kernel<<<NUM_REL / 256, 256, 0, stream>>>(rel_inds, preds_i, out_sop);

  gemm_bias_wmma<<<dim3(HIDDEN / 128, NUM_OBJ / 64), 256, 0, stream>>>(
      obj_fmaps, Wo, bo, objh);
  gemm_bias_wmma<<<dim3(HIDDEN / 128, NUM_REL / 64), 256, 0, stream>>>(
      vr, Wr, br, vrh);

  assemble_kernel<<<NUM_REL, 256, 0, stream>>>(objh, vrh, rel_inds, out_ggnn);
}